// BipartiteEncoder_30176440221877
// MI455X (gfx1250) — compile-verified
//
#include <hip/hip_runtime.h>

// GraphSAGE 2-layer inference for MI455X (gfx1250, wave32).
// Pipeline: degree -> deg_inv -> scatter-mean (L2-resident atomics) -> fused WMMA GEMM (x2).

#define N_NODES 50000
#define E_EDGES 800000
#define IN_DIM  128
#define HID_DIM 256
#define OUT_DIM 128

typedef __attribute__((ext_vector_type(2))) float v2f;
typedef __attribute__((ext_vector_type(8))) float v8f;

// ---------------------------------------------------------------- utilities

__global__ void zero_f32(float* __restrict__ p, long long n) {
  long long i = (long long)blockIdx.x * blockDim.x + threadIdx.x;
  long long stride = (long long)gridDim.x * blockDim.x;
  for (; i < n; i += stride) p[i] = 0.0f;
}

__global__ void degree_kernel(const long long* __restrict__ dst,
                              float* __restrict__ deg, int E) {
  int e = blockIdx.x * blockDim.x + threadIdx.x;
  if (e < E) atomicAdd(&deg[(int)dst[e]], 1.0f);
}

__global__ void deginv_kernel(float* __restrict__ deg, int n) {
  int i = blockIdx.x * blockDim.x + threadIdx.x;
  if (i < n) {
    float d = deg[i];
    deg[i] = (d > 0.0f) ? (1.0f / d) : 0.0f;   // deg is integer-valued >= 1 when > 0
  }
}

// ---------------------------------------------------------- edge scatter-add
// One wave32 per edge. Each lane gathers 4 contiguous channels (float4) from
// the source row and atomically accumulates into the destination row.
// aggr buffers are <= 51 MB -> resident in the 192 MB L2, atomics run at L2 rate.

template<int DIM>
__global__ void scatter_add_kernel(const long long* __restrict__ srcIdx,
                                   const long long* __restrict__ dstIdx,
                                   const float* __restrict__ feat,
                                   float* __restrict__ aggr, int E) {
  const int wid  = (blockIdx.x * blockDim.x + threadIdx.x) >> 5;
  const int lane = threadIdx.x & 31;
  if (wid >= E) return;
  const int s = (int)srcIdx[wid];
  const int d = (int)dstIdx[wid];
  const float4* __restrict__ srow = (const float4*)(feat + (long long)s * DIM);
  float* __restrict__ drow = aggr + (long long)d * DIM;
#pragma unroll
  for (int it = 0; it < DIM / 128; ++it) {
    const int c4 = lane + it * 32;          // float4 index within the row
    float4 v = srow[c4];
    atomicAdd(&drow[c4 * 4 + 0], v.x);
    atomicAdd(&drow[c4 * 4 + 1], v.y);
    atomicAdd(&drow[c4 * 4 + 2], v.z);
    atomicAdd(&drow[c4 * 4 + 3], v.w);
  }
}

// -------------------------------------------------------------- fused GEMM
// Out[n, o] = relu?( sum_k (Anbr[n,k]*dinv[n]) * Wnbr[o,k]
//                  + sum_k  Aroot[n,k]         * Wroot[o,k] + bias[o] )
//
// One wave per 16x(16*NB) output strip: NB accumulators share each A load,
// so per K-step we issue 1 A b64 load + NB B b64 loads for NB WMMAs
// (vs 2 loads per WMMA with a single tile). A-side deg_inv scaling
// (v_pk_mul_f32) is likewise amortized NB-fold.
//
// ISA fp32 operand striping (wave32):
//   A (16x4):  lane L: M = L&15; VGPR0/1 hold K = k0, k0+1 with k0 = (L>>4)*2
//   B (4x16):  lane L: N = L&15; VGPR0/1 hold K = k0, k0+1
//   D (16x16): lane L: N = L&15; VGPR j holds M = j + 8*(L>>4)

template<int K, int ODIM, bool RELU, int NB>
__global__ void sage_gemm_wmma(const float* __restrict__ Anbr,   // [n, K]
                               const float* __restrict__ Aroot,  // [n, K]
                               const float* __restrict__ dinv,   // [n]
                               const float* __restrict__ Wnbr,   // [ODIM, K]
                               const float* __restrict__ Wroot,  // [ODIM, K]
                               const float* __restrict__ bias,   // [ODIM]
                               float* __restrict__ Out,          // [n, ODIM]
                               int nRows) {
  const int lane = threadIdx.x & 31;
  const int wid  = (blockIdx.x * blockDim.x + threadIdx.x) >> 5;
  const int nTilesO = ODIM / (16 * NB);
  const int mTile = wid / nTilesO;
  const int oTile = wid % nTilesO;
  const int mBase = mTile * 16;
  if (mBase >= nRows) return;                 // wave-uniform: EXEC stays all-1s
  const int oBase = oTile * (16 * NB);

  const int mn = lane & 15;                   // M for A-loads, N for B-loads/D
  const int k0 = (lane >> 4) * 2;             // per-lane K pair offset
  const int rowA = (mBase + mn < nRows) ? (mBase + mn) : (nRows - 1);
  const float dscale = dinv[rowA];

  const float* __restrict__ aN = Anbr  + (long long)rowA * K + k0;
  const float* __restrict__ aR = Aroot + (long long)rowA * K + k0;
  const float* bN[NB];
  const float* bR[NB];
#pragma unroll
  for (int nb = 0; nb < NB; ++nb) {
    bN[nb] = Wnbr  + (long long)(oBase + nb * 16 + mn) * K + k0;
    bR[nb] = Wroot + (long long)(oBase + nb * 16 + mn) * K + k0;
  }

  v8f acc[NB];
#pragma unroll
  for (int nb = 0; nb < NB; ++nb) acc[nb] = (v8f){};

#pragma unroll 2
  for (int k = 0; k < K; k += 4) {
    v2f a;
    a[0] = aN[k]     * dscale;                // fold mean-aggregation into A
    a[1] = aN[k + 1] * dscale;
#pragma unroll
    for (int nb = 0; nb < NB; ++nb) {
      v2f b;
      b[0] = bN[nb][k];
      b[1] = bN[nb][k + 1];
      acc[nb] = __builtin_amdgcn_wmma_f32_16x16x4_f32(false, a, false, b,
                                                      (short)0, acc[nb],
                                                      false, false);
    }
  }
#pragma unroll 2
  for (int k = 0; k < K; k += 4) {
    v2f a;
    a[0] = aR[k];
    a[1] = aR[k + 1];
#pragma unroll
    for (int nb = 0; nb < NB; ++nb) {
      v2f b;
      b[0] = bR[nb][k];
      b[1] = bR[nb][k + 1];
      acc[nb] = __builtin_amdgcn_wmma_f32_16x16x4_f32(false, a, false, b,
                                                      (short)0, acc[nb],
                                                      false, false);
    }
  }

  const int mOff = (lane >> 4) * 8;
#pragma unroll
  for (int nb = 0; nb < NB; ++nb) {
    const int col = oBase + nb * 16 + mn;
    const float bval = bias[col];
#pragma unroll
    for (int j = 0; j < 8; ++j) {
      const int row = mBase + mOff + j;
      if (row < nRows) {
        float v = acc[nb][j] + bval;
        if (RELU) v = (v > 0.0f) ? v : 0.0f;
        Out[(long long)row * ODIM + col] = v;
      }
    }
  }
}

// ------------------------------------------------------------------ driver

extern "C" void kernel_launch(void* const* d_in, const int* in_sizes, int n_in,
                              void* d_out, int out_size, void* d_ws, size_t ws_size,
                              hipStream_t stream) {
  (void)in_sizes; (void)n_in; (void)out_size; (void)ws_size;

  const float*     x    = (const float*)d_in[0];
  const long long* eidx = (const long long*)d_in[1];   // int64 [2, E]
  const float*     W1n  = (const float*)d_in[2];       // [HID, IN]
  const float*     W1r  = (const float*)d_in[3];       // [HID, IN]
  const float*     b1   = (const float*)d_in[4];       // [HID]
  const float*     W2n  = (const float*)d_in[5];       // [OUT, HID]
  const float*     W2r  = (const float*)d_in[6];       // [OUT, HID]
  const float*     b2   = (const float*)d_in[7];       // [OUT]
  float* out = (float*)d_out;

  const long long* src = eidx;
  const long long* dst = eidx + E_EDGES;

  // Workspace layout (floats): deg/deg_inv | aggr1 | h1 | aggr2  (~128.2 MB)
  float* deg   = (float*)d_ws;                                   // N
  float* aggr1 = deg   + N_NODES;                                // N*IN
  float* h1    = aggr1 + (long long)N_NODES * IN_DIM;            // N*HID
  float* aggr2 = h1    + (long long)N_NODES * HID_DIM;           // N*HID

  // 1) zero deg + aggr1 (contiguous), compute degree and its inverse
  zero_f32<<<2048, 256, 0, stream>>>(deg, (long long)N_NODES * (1 + IN_DIM));
  degree_kernel<<<(E_EDGES + 255) / 256, 256, 0, stream>>>(dst, deg, E_EDGES);
  deginv_kernel<<<(N_NODES + 255) / 256, 256, 0, stream>>>(deg, N_NODES);

  // 2) layer-1 scatter (x -> aggr1) and fused WMMA GEMM + relu -> h1
  scatter_add_kernel<IN_DIM>
      <<<(E_EDGES * 32 + 255) / 256, 256, 0, stream>>>(src, dst, x, aggr1, E_EDGES);
  {
    const int waves = ((N_NODES + 15) / 16) * (HID_DIM / 64);    // 16x64 strips
    sage_gemm_wmma<IN_DIM, HID_DIM, true, 4>
        <<<(waves * 32 + 255) / 256, 256, 0, stream>>>(aggr1, x, deg,
                                                       W1n, W1r, b1, h1, N_NODES);
  }

  // 3) layer-2 scatter (h1 -> aggr2) and fused WMMA GEMM -> out
  zero_f32<<<2048, 256, 0, stream>>>(aggr2, (long long)N_NODES * HID_DIM);
  scatter_add_kernel<HID_DIM>
      <<<(E_EDGES * 32 + 255) / 256, 256, 0, stream>>>(src, dst, h1, aggr2, E_EDGES);
  {
    const int waves = ((N_NODES + 15) / 16) * (OUT_DIM / 64);    // 16x64 strips
    sage_gemm_wmma<HID_DIM, OUT_DIM, false, 4>
        <<<(waves * 32 + 255) / 256, 256, 0, stream>>>(aggr2, h1, deg,
                                                       W2n, W2r, b2, out, N_NODES);
  }
}